// CG_46617575031388
// MI455X (gfx1250) — compile-verified
//
#include <hip/hip_runtime.h>

#define N 2048
#define BATCH 8
#define CG_TOL 1e-5f
#define MAXITER 64

typedef __attribute__((ext_vector_type(2))) float v2f;
typedef __attribute__((ext_vector_type(8))) float v8f;

// ---------------------------------------------------------------------------
// init: x = 0, r = b, p = 0
__global__ void k_init_vec(const float* __restrict__ b, float* __restrict__ x,
                           float* __restrict__ r, float* __restrict__ p) {
    int i = blockIdx.x * blockDim.x + threadIdx.x;   // 16384 total
    float v = b[i];
    x[i] = 0.f;
    r[i] = v;
    p[i] = 0.f;
}

// init: atol2[b] = TOL^2 * ||b||^2 ; seed rr_part with ||r0||^2 = ||b||^2
__global__ void k_init_red(const float* __restrict__ b, float* __restrict__ atol2,
                           float* __restrict__ rr_part, float* __restrict__ rho_prev,
                           int* __restrict__ done) {
    __shared__ float s[256];
    int bb = blockIdx.x;                 // one block per batch
    const float* bp = b + bb * N;
    float acc = 0.f;
    for (int i = threadIdx.x; i < N; i += 256) { float t = bp[i]; acc += t * t; }
    s[threadIdx.x] = acc;
    __syncthreads();
    for (int off = 128; off > 0; off >>= 1) {
        if (threadIdx.x < off) s[threadIdx.x] += s[threadIdx.x + off];
        __syncthreads();
    }
    if (threadIdx.x == 0) {
        float ss = s[0];
        atol2[bb] = CG_TOL * CG_TOL * ss;
        rr_part[bb * 8 + 0] = ss;
        rho_prev[bb] = 1.f;
        if (bb == 0) *done = 0;
    }
    if (threadIdx.x < 7) rr_part[bb * 8 + 1 + threadIdx.x] = 0.f;
}

// per-iteration scalars: rho, done check (freeze), beta, rho_prev
__global__ void k_scalars(const float* __restrict__ rr_part, const float* __restrict__ atol2,
                          float* __restrict__ rho, float* __restrict__ rho_prev,
                          float* __restrict__ beta, int* __restrict__ done, int it) {
    __shared__ float srho[BATCH];
    __shared__ int alldone;
    int t = threadIdx.x;                 // 32 threads
    if (t < BATCH) {
        float s = 0.f;
        for (int i = 0; i < 8; i++) s += rr_part[t * 8 + i];
        srho[t] = s;
    }
    __syncthreads();
    if (t == 0) {
        int d = *done;
        if (!d) {
            bool all_lt = true, all_zero = true;
            for (int i = 0; i < BATCH; i++) {
                if (!(srho[i] < atol2[i])) all_lt = false;
                if (srho[i] != 0.f)        all_zero = false;
            }
            if (all_lt || (it == 0 && all_zero)) { d = 1; *done = 1; }
        }
        alldone = d;
    }
    __syncthreads();
    if (alldone) return;
    if (t < BATCH) {
        rho[t]      = srho[t];
        beta[t]     = (it > 0) ? srho[t] / rho_prev[t] : 0.f;
        rho_prev[t] = srho[t];
    }
}

// p = r + beta * p
__global__ void k_axpy_p(const float* __restrict__ r, float* __restrict__ p,
                         const float* __restrict__ beta, const int* __restrict__ done) {
    if (*done) return;
    int i = blockIdx.x * 256 + threadIdx.x;
    int bb = i >> 11;
    p[i] = r[i] + beta[bb] * p[i];
}

// q = A @ p via V_WMMA_F32_16X16X4_F32; also per-block partial of p·q.
// Grid: 128 blocks * 256 threads. 16 blocks/batch; 8 waves/block, 1 stripe (16 rows)/wave.
__global__ void __launch_bounds__(256) k_matvec(const float* __restrict__ A,
                                                const float* __restrict__ p,
                                                float* __restrict__ q,
                                                float* __restrict__ pq_part,
                                                const int* __restrict__ done) {
    if (*done) return;
    __shared__ __align__(16) float sp[N];
    __shared__ float red[16];
    const int bb     = blockIdx.x >> 4;
    const int blkInB = blockIdx.x & 15;
    const int wave   = threadIdx.x >> 5;
    const int lane   = threadIdx.x & 31;
    const int h      = lane >> 4;        // half: selects K-pair / column group
    const int m      = lane & 15;        // row within stripe

    // stage p[bb] into LDS (shared by the 8 waves)
    {
        const float4* ps = (const float4*)(p + bb * N);
        float4* sd = (float4*)sp;
        for (int i = threadIdx.x; i < N / 4; i += 256) sd[i] = ps[i];
    }
    __syncthreads();

    const int row0 = blkInB * 128 + wave * 16;
    const float* Arow = A + ((size_t)bb * N + (size_t)(row0 + m)) * N;

    v8f acc = {0.f, 0.f, 0.f, 0.f, 0.f, 0.f, 0.f, 0.f};
    for (int j0 = 0; j0 < N; j0 += 16) {
        // each lane: 32B of its A row (cols j0+8h .. j0+8h+7), matching p values
        const float4 a0 = *(const float4*)(Arow + j0 + 8 * h);
        const float4 a1 = *(const float4*)(Arow + j0 + 8 * h + 4);
        const float4 p0 = *(const float4*)(sp + j0 + 8 * h);
        const float4 p1 = *(const float4*)(sp + j0 + 8 * h + 4);
        v2f av, bv;
        av.x = a0.x; av.y = a0.y; bv.x = p0.x; bv.y = p0.y;
        acc = __builtin_amdgcn_wmma_f32_16x16x4_f32(false, av, false, bv, (short)0, acc, false, false);
        av.x = a0.z; av.y = a0.w; bv.x = p0.z; bv.y = p0.w;
        acc = __builtin_amdgcn_wmma_f32_16x16x4_f32(false, av, false, bv, (short)0, acc, false, false);
        av.x = a1.x; av.y = a1.y; bv.x = p1.x; bv.y = p1.y;
        acc = __builtin_amdgcn_wmma_f32_16x16x4_f32(false, av, false, bv, (short)0, acc, false, false);
        av.x = a1.z; av.y = a1.w; bv.x = p1.z; bv.y = p1.w;
        acc = __builtin_amdgcn_wmma_f32_16x16x4_f32(false, av, false, bv, (short)0, acc, false, false);
    }

    // D[m,n] is independent of n: lane n==0 of each half holds q[row0+8h+v] in acc[v]
    if (m == 0) {
        float* qp = q + bb * N + row0 + 8 * h;
        float pqs = 0.f;
        #pragma unroll
        for (int v = 0; v < 8; v++) {
            qp[v] = acc[v];
            pqs += acc[v] * sp[row0 + 8 * h + v];
        }
        red[wave * 2 + h] = pqs;
    }
    __syncthreads();
    if (threadIdx.x == 0) {
        float s = 0.f;
        for (int k = 0; k < 16; k++) s += red[k];
        pq_part[bb * 16 + blkInB] = s;
    }
}

// alpha = rho/pq ; x += alpha p ; r -= alpha q ; partial ||r_new||^2 per block
__global__ void k_update(float* __restrict__ x, float* __restrict__ r,
                         const float* __restrict__ p, const float* __restrict__ q,
                         const float* __restrict__ rho, const float* __restrict__ pq_part,
                         float* __restrict__ rr_part, const int* __restrict__ done) {
    if (*done) return;
    __shared__ float s[256];
    int i  = blockIdx.x * 256 + threadIdx.x;   // 64 blocks, 8 per batch
    int bb = i >> 11;
    int chunk = blockIdx.x & 7;
    float pqs = 0.f;
    for (int k = 0; k < 16; k++) pqs += pq_part[bb * 16 + k];  // fixed-order, deterministic
    float alpha = rho[bb] / pqs;
    float pi = p[i], qi = q[i];
    x[i] += alpha * pi;
    float rn = r[i] - alpha * qi;
    r[i] = rn;
    s[threadIdx.x] = rn * rn;
    __syncthreads();
    for (int off = 128; off > 0; off >>= 1) {
        if (threadIdx.x < off) s[threadIdx.x] += s[threadIdx.x + off];
        __syncthreads();
    }
    if (threadIdx.x == 0) rr_part[bb * 8 + chunk] = s[0];
}

// ---------------------------------------------------------------------------
extern "C" void kernel_launch(void* const* d_in, const int* in_sizes, int n_in,
                              void* d_out, int out_size, void* d_ws, size_t ws_size,
                              hipStream_t stream) {
    const float* A = (const float*)d_in[0];   // [8,2048,2048]
    const float* b = (const float*)d_in[1];   // [8,2048]
    float* x = (float*)d_out;                 // [8,2048]

    float* ws = (float*)d_ws;
    float* r        = ws;                       // 16384
    float* p        = ws + 16384;               // 16384
    float* q        = ws + 32768;               // 16384
    float* rho      = ws + 49152;               // 8
    float* rho_prev = ws + 49160;               // 8
    float* beta     = ws + 49168;               // 8
    float* atol2    = ws + 49176;               // 8
    float* rr_part  = ws + 49184;               // 64
    float* pq_part  = ws + 49248;               // 128
    int*   done     = (int*)(ws + 49376);       // 1

    k_init_vec<<<64, 256, 0, stream>>>(b, x, r, p);
    k_init_red<<<BATCH, 256, 0, stream>>>(b, atol2, rr_part, rho_prev, done);

    for (int it = 0; it < MAXITER; ++it) {
        k_scalars<<<1, 32, 0, stream>>>(rr_part, atol2, rho, rho_prev, beta, done, it);
        k_axpy_p<<<64, 256, 0, stream>>>(r, p, beta, done);
        k_matvec<<<128, 256, 0, stream>>>(A, p, q, pq_part, done);
        k_update<<<64, 256, 0, stream>>>(x, r, p, q, rho, pq_part, rr_part, done);
    }
}